// GNNattn_26087631356356
// MI455X (gfx1250) — compile-verified
//
#include <hip/hip_runtime.h>

#define N_NODES 50000
#define N_EDGES 300000
#define DHID    128     // hidden / head_dim
#define FQKV    256     // HEADS * DHID
#define MTILES  (N_NODES / 16)        // 3125
#define MGROUPS ((MTILES + 7) / 8)    // 391 (8 m-tiles per block)
#define LDS_STRIDE 132                // 128 + 4 pad: bank = 4*c + k (mod 64)

typedef __attribute__((ext_vector_type(2))) float v2f;
typedef __attribute__((ext_vector_type(8))) float v8f;

#if __has_builtin(__builtin_amdgcn_wmma_f32_16x16x4_f32)
#define HAVE_WMMA_F32 1
#endif

// ---------------------------------------------------------------------------
// GEMM + bias: Y[rows,Fout] = X[rows,128] @ W[128,Fout] + b[Fout]
// Block = 256 threads (8 waves) = 8 M-tiles x one 64-col N panel.
// W panel staged transposed in LDS: Wt[c][k] so a B fragment (k,k+1 pair)
// is one aligned ds_load_b64. Per wave K-step: 1 global b64 (A) +
// 4 ds b64 (B) + 4 v_wmma_f32_16x16x4_f32.
// ---------------------------------------------------------------------------
__global__ __launch_bounds__(256) void gemm_bias_kernel(
    const float* __restrict__ X, const float* __restrict__ W,
    const float* __restrict__ bias, float* __restrict__ Y, int Fout)
{
    __shared__ float smem[64 * LDS_STRIDE];   // 33792 B

    const int nt0 = blockIdx.y * 64;          // panel's first output column

    // cooperative load of W[:, nt0:nt0+64], transposed into LDS
    for (int i = threadIdx.x; i < 64 * 128; i += 256) {
        const int kidx = i >> 6;              // 0..127
        const int c    = i & 63;              // 0..63 (coalesced global read)
        smem[c * LDS_STRIDE + kidx] = W[(size_t)kidx * Fout + nt0 + c];
    }
    __syncthreads();

    const int lane  = threadIdx.x & 31;
    const int wave  = threadIdx.x >> 5;
    const int mt    = blockIdx.x * 8 + wave;  // 16-row tile index
    if (mt >= MTILES) return;                 // whole-wave exit (after barrier)

    const int m_in  = lane & 15;
    const int khalf = lane >> 4;              // 0 or 1 (K half select)

    // A fragment source: X[mt*16 + m_in][kk + 2*khalf + {0,1}]
    const float* xrow = X + (size_t)(mt * 16 + m_in) * DHID + 2 * khalf;

    v8f acc[4] = {};

#ifdef HAVE_WMMA_F32
    #pragma unroll 4
    for (int kk = 0; kk < DHID; kk += 4) {
        float2 av = *(const float2*)(xrow + kk);
        v2f a; a.x = av.x; a.y = av.y;
        const int krow = kk + 2 * khalf;
        #pragma unroll
        for (int t = 0; t < 4; ++t) {
            float2 bv = *(const float2*)&smem[(t * 16 + m_in) * LDS_STRIDE + krow];
            v2f b; b.x = bv.x; b.y = bv.y;
            acc[t] = __builtin_amdgcn_wmma_f32_16x16x4_f32(
                false, a, false, b, (short)0, acc[t], false, false);
        }
    }
#else
    // Scalar fallback producing the identical register layout.
    for (int kk = 0; kk < DHID; ++kk) {
        #pragma unroll
        for (int t = 0; t < 4; ++t) {
            float wv = smem[(t * 16 + m_in) * LDS_STRIDE + kk];
            #pragma unroll
            for (int r = 0; r < 8; ++r)
                acc[t][r] += X[(size_t)(mt * 16 + 8 * khalf + r) * DHID + kk] * wv;
        }
    }
#endif

    // C/D layout: VGPR r -> row = r + 8*khalf ; col = lane&15 within sub-tile
    const int rbase = mt * 16 + 8 * khalf;
    #pragma unroll
    for (int t = 0; t < 4; ++t) {
        const int col = nt0 + t * 16 + m_in;
        const float bv = bias[col];
        #pragma unroll
        for (int r = 0; r < 8; ++r)
            Y[(size_t)(rbase + r) * Fout + col] = acc[t][r] + bv;
    }
}

// ---------------------------------------------------------------------------
// Monotonic float<->uint encoding for atomic segment-max.
// ---------------------------------------------------------------------------
__device__ __forceinline__ unsigned encf(float f) {
    unsigned u = __float_as_uint(f);
    return (u & 0x80000000u) ? ~u : (u | 0x80000000u);
}
__device__ __forceinline__ float decf(unsigned e) {
    unsigned u = (e & 0x80000000u) ? (e & 0x7FFFFFFFu) : ~e;
    return __uint_as_float(u);
}

// ---------------------------------------------------------------------------
// E1: one wave per edge. alpha[e,h] = dot(q[dst,h,:], k[src,h,:]) / sqrt(128)
//     + atomic segment max into menc[dst,h].
// ---------------------------------------------------------------------------
__global__ __launch_bounds__(256) void edge_alpha_max_kernel(
    const float* __restrict__ q, const float* __restrict__ k,
    const int* __restrict__ src, const int* __restrict__ dst,
    float* __restrict__ alpha, unsigned* __restrict__ menc)
{
    const int e    = (blockIdx.x * blockDim.x + threadIdx.x) >> 5;
    const int lane = threadIdx.x & 31;
    if (e >= N_EDGES) return;
    const int s = src[e], d = dst[e];
    const float4* qp = (const float4*)(q + (size_t)d * FQKV);
    const float4* kp = (const float4*)(k + (size_t)s * FQKV);
    #pragma unroll
    for (int h = 0; h < 2; ++h) {
        float4 qa = qp[h * 32 + lane];
        float4 ka = kp[h * 32 + lane];
        float p = qa.x * ka.x + qa.y * ka.y + qa.z * ka.z + qa.w * ka.w;
        #pragma unroll
        for (int off = 16; off > 0; off >>= 1)
            p += __shfl_xor(p, off, 32);
        if (lane == 0) {
            float a = p * 0.08838834764831845f;   // 1/sqrt(128)
            alpha[(size_t)e * 2 + h] = a;
            atomicMax(&menc[(size_t)d * 2 + h], encf(a));
        }
    }
}

// ---------------------------------------------------------------------------
// E2: one thread per (edge, head): ex = exp(alpha - max); denom += ex.
// ---------------------------------------------------------------------------
__global__ __launch_bounds__(256) void edge_exp_kernel(
    float* __restrict__ alpha, const unsigned* __restrict__ menc,
    float* __restrict__ denom, const int* __restrict__ dst)
{
    const int i = blockIdx.x * blockDim.x + threadIdx.x;
    if (i >= N_EDGES * 2) return;
    const int e = i >> 1, h = i & 1;
    const int d = dst[e];
    float m  = decf(menc[(size_t)d * 2 + h]);
    float ex = expf(alpha[i] - m);
    alpha[i] = ex;                         // ex stored in place
    atomicAdd(&denom[(size_t)d * 2 + h], ex);
}

// ---------------------------------------------------------------------------
// E3: one wave per edge: agg[dst,h,:] += (ex / (denom+eps)) * v[src,h,:]
// ---------------------------------------------------------------------------
__global__ __launch_bounds__(256) void edge_agg_kernel(
    const float* __restrict__ ex, const float* __restrict__ denom,
    const float* __restrict__ v, const int* __restrict__ src,
    const int* __restrict__ dst, float* __restrict__ agg)
{
    const int e    = (blockIdx.x * blockDim.x + threadIdx.x) >> 5;
    const int lane = threadIdx.x & 31;
    if (e >= N_EDGES) return;
    const int s = src[e], d = dst[e];
    const float4* vp = (const float4*)(v + (size_t)s * FQKV);
    #pragma unroll
    for (int h = 0; h < 2; ++h) {
        float w = ex[(size_t)e * 2 + h] / (denom[(size_t)d * 2 + h] + 1e-16f);
        float4 va = vp[h * 32 + lane];
        float* ap = agg + (size_t)d * FQKV + h * DHID + lane * 4;
        atomicAdd(ap + 0, w * va.x);
        atomicAdd(ap + 1, w * va.y);
        atomicAdd(ap + 2, w * va.z);
        atomicAdd(ap + 3, w * va.w);
    }
}

// ---------------------------------------------------------------------------
// N1: out[n,:] = 0.5*(agg[n,0,:] + agg[n,1,:]) + skip[n,:] + lin[n,:]
// ---------------------------------------------------------------------------
__global__ __launch_bounds__(256) void combine_kernel(
    const float* __restrict__ agg, const float* __restrict__ skip,
    const float* __restrict__ lin, float* __restrict__ out)
{
    const int i = blockIdx.x * blockDim.x + threadIdx.x;
    if (i >= N_NODES * DHID) return;
    const int n = i >> 7, d = i & 127;
    const size_t a0 = (size_t)n * FQKV + d;
    out[i] = 0.5f * (agg[a0] + agg[a0 + DHID]) + skip[i] + lin[i];
}

// ---------------------------------------------------------------------------
extern "C" void kernel_launch(void* const* d_in, const int* in_sizes, int n_in,
                              void* d_out, int out_size, void* d_ws, size_t ws_size,
                              hipStream_t stream)
{
    (void)in_sizes; (void)n_in; (void)out_size; (void)ws_size;

    const float* x   = (const float*)d_in[0];
    const int*   ei  = (const int*)d_in[1];
    const int*   src = ei;
    const int*   dst = ei + N_EDGES;

    // layer weight/bias pointers: {layer0, layer1}
    const float* qw[2] = {(const float*)d_in[2],  (const float*)d_in[10]};
    const float* qb[2] = {(const float*)d_in[3],  (const float*)d_in[11]};
    const float* kw[2] = {(const float*)d_in[4],  (const float*)d_in[12]};
    const float* kb[2] = {(const float*)d_in[5],  (const float*)d_in[13]};
    const float* vw[2] = {(const float*)d_in[6],  (const float*)d_in[14]};
    const float* vb[2] = {(const float*)d_in[7],  (const float*)d_in[15]};
    const float* sw[2] = {(const float*)d_in[8],  (const float*)d_in[16]};
    const float* sb[2] = {(const float*)d_in[9],  (const float*)d_in[17]};
    const float* lw[2] = {(const float*)d_in[18], (const float*)d_in[20]};
    const float* lb[2] = {(const float*)d_in[19], (const float*)d_in[21]};

    // workspace layout (floats)
    float* ws    = (float*)d_ws;
    float* qbuf  = ws;                                  // N*256
    float* kbuf  = qbuf  + (size_t)N_NODES * FQKV;      // N*256
    float* vbuf  = kbuf  + (size_t)N_NODES * FQKV;      // N*256
    float* skipb = vbuf  + (size_t)N_NODES * FQKV;      // N*128
    float* linb  = skipb + (size_t)N_NODES * DHID;      // N*128
    float* hbuf  = linb  + (size_t)N_NODES * DHID;      // N*128
    float* alphb = hbuf  + (size_t)N_NODES * DHID;      // E*2 (alpha, then ex)
    float* denom = alphb + (size_t)N_EDGES * 2;         // N*2
    unsigned* menc = (unsigned*)(denom + (size_t)N_NODES * 2);  // N*2
    float* aggb  = qbuf;  // alias: q dead after E1

    const int eblocks  = N_EDGES / 8;                   // wave/edge, 256 thr
    const int e2blocks = (N_EDGES * 2 + 255) / 256;
    const int cblocks  = (N_NODES * DHID + 255) / 256;

    const dim3 gqkv(MGROUPS, FQKV / 64);                // 391 x 4
    const dim3 gsml(MGROUPS, DHID / 64);                // 391 x 2

    for (int l = 0; l < 2; ++l) {
        const float* in_x = (l == 0) ? x : hbuf;
        float* outp       = (l == 0) ? hbuf : (float*)d_out;

        // dense projections
        gemm_bias_kernel<<<gqkv, 256, 0, stream>>>(in_x, qw[l], qb[l], qbuf, FQKV);
        gemm_bias_kernel<<<gqkv, 256, 0, stream>>>(in_x, kw[l], kb[l], kbuf, FQKV);
        gemm_bias_kernel<<<gqkv, 256, 0, stream>>>(in_x, vw[l], vb[l], vbuf, FQKV);
        gemm_bias_kernel<<<gsml, 256, 0, stream>>>(in_x, sw[l], sb[l], skipb, DHID);
        gemm_bias_kernel<<<gsml, 256, 0, stream>>>(in_x, lw[l], lb[l], linb, DHID);

        // segment softmax state
        hipMemsetAsync(denom, 0, (size_t)N_NODES * 2 * sizeof(float), stream);
        hipMemsetAsync(menc,  0, (size_t)N_NODES * 2 * sizeof(unsigned), stream);

        edge_alpha_max_kernel<<<eblocks, 256, 0, stream>>>(
            qbuf, kbuf, src, dst, alphb, menc);

        // q dead now -> reuse as agg
        hipMemsetAsync(aggb, 0, (size_t)N_NODES * FQKV * sizeof(float), stream);

        edge_exp_kernel<<<e2blocks, 256, 0, stream>>>(alphb, menc, denom, dst);
        edge_agg_kernel<<<eblocks, 256, 0, stream>>>(
            alphb, denom, vbuf, src, dst, aggb);

        combine_kernel<<<cblocks, 256, 0, stream>>>(aggb, skipb, linb, outp);
    }
}